// EaDA_29386166239699
// MI455X (gfx1250) — compile-verified
//
#include <hip/hip_runtime.h>
#include <hip/hip_bf16.h>

#define D128 128
#define C256 256
#define NT 10
#define GAMMA 0.4f
#define BN_EPS 1e-5f

typedef __attribute__((ext_vector_type(2))) float v2f;
typedef __attribute__((ext_vector_type(8))) float v8f;

// ---------------------------------------------------------------- node encode
__global__ void k_nodeenc(const float* __restrict__ x, const float* __restrict__ W,
                          const float* __restrict__ b, float* __restrict__ h0, int total) {
  int i = blockIdx.x * blockDim.x + threadIdx.x;
  if (i >= total) return;
  int n = i >> 7, d = i & 127;
  const float* xr = x + (size_t)n * 4;
  float s = b[d];
  s += xr[0] * W[0 * D128 + d];
  s += xr[1] * W[1 * D128 + d];
  s += xr[2] * W[2 * D128 + d];
  s += xr[3] * W[3 * D128 + d];
  h0[i] = s;
}

// ------------------------------------------------- edge message + scatter add
__global__ void k_edge(const float* __restrict__ h, const float* __restrict__ emb,
                       const int* __restrict__ src, const int* __restrict__ dst,
                       const int* __restrict__ ea, float* __restrict__ aggr, int E) {
  int wave = threadIdx.x >> 5, lane = threadIdx.x & 31;
  int e = blockIdx.x * 8 + wave;
  if (e >= E) return;
  int s = src[e], d = dst[e], a = ea[e];
  const float4 hv = *(const float4*)(h + (size_t)s * D128 + lane * 4);
  const float4 ev = *(const float4*)(emb + (size_t)a * D128 + lane * 4);
  float4 m;
  m.x = fmaxf(hv.x + ev.x, 0.f);
  m.y = fmaxf(hv.y + ev.y, 0.f);
  m.z = fmaxf(hv.z + ev.z, 0.f);
  m.w = fmaxf(hv.w + ev.w, 0.f);
  float* o = aggr + (size_t)d * D128 + lane * 4;
  unsafeAtomicAdd(o + 0, m.x);
  unsafeAtomicAdd(o + 1, m.y);
  unsafeAtomicAdd(o + 2, m.z);
  unsafeAtomicAdd(o + 3, m.w);
}

// -------------------------------------------------- z = (1+eps)*h + aggr (in place)
__global__ void k_zbuild(float* __restrict__ ag, const float* __restrict__ h,
                         const float* __restrict__ epsp, int total) {
  int i = blockIdx.x * blockDim.x + threadIdx.x;
  if (i >= total) return;
  ag[i] = (1.f + epsp[0]) * h[i] + ag[i];
}

// ---------------------------------------------------------------- WMMA GEMM
// C[M,NC] = act(A[M,K]) @ W[K,NC] + bias ; act = (BN? relu(a*scale+shift) : a)
// One wave computes a 16x64 output tile: a single fp32 A fragment
// (V_WMMA_F32_16X16X4_F32 layout: lanes 0-15/16-31 both hold M=0..15, lane-half
// selects K+0/K+2, vector elem selects K parity) feeds 4 accumulators, so A
// traffic is amortized 4x and the inner loop is 4 wmma : 10 loads.
template <int BN>
__global__ __launch_bounds__(256) void k_gemm(const float* __restrict__ A,
    const float* __restrict__ W, const float* __restrict__ bias, float* __restrict__ C,
    int M, int K, int NC, const float* __restrict__ scale, const float* __restrict__ shift) {
  int wave = threadIdx.x >> 5, lane = threadIdx.x & 31;
  int grp = NC >> 6;                       // groups of 64 columns
  int tile = blockIdx.x * 8 + wave;
  int totTiles = (M >> 4) * grp;
  if (tile >= totTiles) return;            // wave-uniform exit -> EXEC stays all-ones
  int tm = (tile / grp) << 4;
  int tn = (tile % grp) << 6;
  int half = lane >> 4;
  int l15 = lane & 15;
  const float* arow = A + (size_t)(tm + l15) * K;
  int n0 = tn + l15;
  v8f acc0 = {}, acc1 = {}, acc2 = {}, acc3 = {};
  for (int k = 0; k < K; k += 4) {
    int ka = k + 2 * half;
    v2f a;
    if (BN) {
      float a0 = fmaf(arow[ka], scale[ka], shift[ka]);
      float a1 = fmaf(arow[ka + 1], scale[ka + 1], shift[ka + 1]);
      a[0] = fmaxf(a0, 0.f);
      a[1] = fmaxf(a1, 0.f);
    } else {
      a[0] = arow[ka];
      a[1] = arow[ka + 1];
    }
    const float* w0 = W + (size_t)ka * NC + n0;
    const float* w1 = W + (size_t)(ka + 1) * NC + n0;
    v2f b;
    b[0] = w0[0];  b[1] = w1[0];
    acc0 = __builtin_amdgcn_wmma_f32_16x16x4_f32(false, a, false, b, (short)0, acc0, false, false);
    b[0] = w0[16]; b[1] = w1[16];
    acc1 = __builtin_amdgcn_wmma_f32_16x16x4_f32(false, a, false, b, (short)0, acc1, false, false);
    b[0] = w0[32]; b[1] = w1[32];
    acc2 = __builtin_amdgcn_wmma_f32_16x16x4_f32(false, a, false, b, (short)0, acc2, false, false);
    b[0] = w0[48]; b[1] = w1[48];
    acc3 = __builtin_amdgcn_wmma_f32_16x16x4_f32(false, a, false, b, (short)0, acc3, false, false);
  }
  float bv0 = bias ? bias[n0 + 0]  : 0.f;
  float bv1 = bias ? bias[n0 + 16] : 0.f;
  float bv2 = bias ? bias[n0 + 32] : 0.f;
  float bv3 = bias ? bias[n0 + 48] : 0.f;
#pragma unroll
  for (int r = 0; r < 8; ++r) {
    int row = tm + r + half * 8;           // C layout: VGPR r -> M = r + 8*half
    float* crow = C + (size_t)row * NC + n0;
    crow[0]  = acc0[r] + bv0;
    crow[16] = acc1[r] + bv1;
    crow[32] = acc2[r] + bv2;
    crow[48] = acc3[r] + bv3;
  }
}

// --------------------------------------------------------- BN column statistics
__global__ void k_stats(const float* __restrict__ X, int rows, int cols,
                        float* __restrict__ stats) {
  int c = threadIdx.x;                       // blockDim.x == cols
  int chunk = (rows + gridDim.x - 1) / gridDim.x;
  int r0 = blockIdx.x * chunk;
  int r1 = min(r0 + chunk, rows);
  float s = 0.f, q = 0.f;
  for (int r = r0; r < r1; ++r) {
    float v = X[(size_t)r * cols + c];
    s += v;
    q += v * v;
  }
  unsafeAtomicAdd(&stats[c], s);
  unsafeAtomicAdd(&stats[cols + c], q);
}

__global__ void k_bnfin(const float* __restrict__ stats, const float* __restrict__ g,
                        const float* __restrict__ b, int rows, int cols,
                        float* __restrict__ scale, float* __restrict__ shift) {
  int c = threadIdx.x;
  if (c >= cols) return;
  float m = stats[c] / (float)rows;
  float v = stats[cols + c] / (float)rows - m * m;
  float s = g[c] * rsqrtf(v + BN_EPS);
  scale[c] = s;
  shift[c] = b[c] - m * s;
}

// -------------------------------------------------- h += [relu](bn(z))   (residual)
__global__ void k_resid(float* __restrict__ h, const float* __restrict__ z,
                        const float* __restrict__ scale, const float* __restrict__ shift,
                        int doRelu, int total) {
  int i = blockIdx.x * blockDim.x + threadIdx.x;
  if (i >= total) return;
  int c = i & 127;
  float v = fmaf(z[i], scale[c], shift[c]);
  if (doRelu) v = fmaxf(v, 0.f);
  h[i] += v;
}

// ------------------------------------------------------- gate head (logits->binary)
__global__ void k_gate(const float* __restrict__ t1, const float* __restrict__ sc,
                       const float* __restrict__ sh, const float* __restrict__ W2,
                       const float* __restrict__ b2, const float* __restrict__ gum,
                       float* __restrict__ gate, int N) {
  int wave = threadIdx.x >> 5, lane = threadIdx.x & 31;
  int n = blockIdx.x * 8 + wave;
  if (n >= N) return;
  const float* row = t1 + (size_t)n * C256;
  float l0 = 0.f, l1 = 0.f;
  for (int k = lane; k < C256; k += 32) {
    float v = fmaxf(fmaf(row[k], sc[k], sh[k]), 0.f);
    l0 += v * W2[k * 2 + 0];
    l1 += v * W2[k * 2 + 1];
  }
  for (int o = 16; o; o >>= 1) {
    l0 += __shfl_xor(l0, o, 32);
    l1 += __shfl_xor(l1, o, 32);
  }
  if (lane == 0) {
    float y0 = l0 + b2[0] + gum[(size_t)n * 2 + 0];
    float y1 = l1 + b2[1] + gum[(size_t)n * 2 + 1];
    gate[n] = (y1 > y0) ? 1.f : 0.f;   // argmax ties -> index 0 (gate 0), matches jnp.argmax
  }
}

// ------------------------------------------------------------- pooling (scatter)
__global__ void k_pool(const float* __restrict__ hN, const float* __restrict__ gate,
                       const int* __restrict__ batch, float* __restrict__ poolA,
                       float* __restrict__ cnts, float* __restrict__ rn,
                       float* __restrict__ en, int N) {
  int wave = threadIdx.x >> 5, lane = threadIdx.x & 31;
  int n = blockIdx.x * 8 + wave;
  if (n >= N) return;
  int b = batch[n];
  float g = gate[n];
  if (lane == 0) {
    unsafeAtomicAdd(&cnts[b], 1.f);
    unsafeAtomicAdd(&rn[b], g);
    unsafeAtomicAdd(&en[b], 1.f - g);
  }
  if (g != 0.f) {
    const float4 v = *(const float4*)(hN + (size_t)n * D128 + lane * 4);
    float* o = poolA + (size_t)b * D128 + lane * 4;
    unsafeAtomicAdd(o + 0, v.x * g);
    unsafeAtomicAdd(o + 1, v.y * g);
    unsafeAtomicAdd(o + 2, v.z * g);
    unsafeAtomicAdd(o + 3, v.w * g);
  }
}

__global__ void k_poolfin(float* __restrict__ poolA, const float* __restrict__ cnts, int total) {
  int i = blockIdx.x * blockDim.x + threadIdx.x;
  if (i >= total) return;
  poolA[i] /= fmaxf(cnts[i >> 7], 1.f);
}

// ---------------------------------------------------------- small output head
__global__ void k_head(const float* __restrict__ tp, const float* __restrict__ sc,
                       const float* __restrict__ sh, const float* __restrict__ W2,
                       const float* __restrict__ b2, float* __restrict__ out,
                       int B, int K, int OC) {
  int i = blockIdx.x * blockDim.x + threadIdx.x;
  if (i >= B * OC) return;
  int r = i / OC, j = i % OC;
  const float* row = tp + (size_t)r * K;
  float s = b2[j];
  for (int k = 0; k < K; ++k) {
    float v = fmaxf(fmaf(row[k], sc[k], sh[k]), 0.f);
    s += v * W2[k * OC + j];
  }
  out[i] = s;
}

// --------------------------------------------------------------------- loss
__global__ void k_loss(const float* __restrict__ rn, const float* __restrict__ en,
                       float* __restrict__ out, int B) {
  __shared__ float sm[256];
  float s = 0.f;
  for (int b = threadIdx.x; b < B; b += 256) {
    float r = rn[b] + 1e-8f, e = en[b] + 1e-8f;
    s += fabsf(r / (r + e) - GAMMA);
  }
  sm[threadIdx.x] = s;
  __syncthreads();
  for (int o = 128; o; o >>= 1) {
    if (threadIdx.x < o) sm[threadIdx.x] += sm[threadIdx.x + o];
    __syncthreads();
  }
  if (threadIdx.x == 0) out[0] = sm[0] / (float)B;
}

// ============================================================================
extern "C" void kernel_launch(void* const* d_in, const int* in_sizes, int n_in,
                              void* d_out, int out_size, void* d_ws, size_t ws_size,
                              hipStream_t stream) {
  const int N = in_sizes[0] / 4;
  const int E = in_sizes[2];
  const int B = (out_size - 1) / NT;

  const float* x       = (const float*)d_in[0];
  const int*   src     = (const int*)d_in[1];
  const int*   dst     = src + E;
  const int*   ea      = (const int*)d_in[2];
  const int*   batch   = (const int*)d_in[3];
  const float* gumbels = (const float*)d_in[5];
  const float* node_W  = (const float*)d_in[6];
  const float* node_b  = (const float*)d_in[7];
  const float* enc_eps = (const float*)d_in[8];
  const float* enc_edge= (const float*)d_in[9];
  const float* enc_W1  = (const float*)d_in[10];
  const float* enc_b1  = (const float*)d_in[11];
  const float* enc_g1  = (const float*)d_in[12];
  const float* enc_bb1 = (const float*)d_in[13];
  const float* enc_W2  = (const float*)d_in[14];
  const float* enc_b2  = (const float*)d_in[15];
  const float* enc_g2  = (const float*)d_in[16];
  const float* enc_bb2 = (const float*)d_in[17];
  const float* rat_eps = (const float*)d_in[18];
  const float* rat_edge= (const float*)d_in[19];
  const float* rat_W1  = (const float*)d_in[20];
  const float* rat_b1  = (const float*)d_in[21];
  const float* rat_g1  = (const float*)d_in[22];
  const float* rat_bb1 = (const float*)d_in[23];
  const float* rat_W2  = (const float*)d_in[24];
  const float* rat_b2  = (const float*)d_in[25];
  const float* rat_g2  = (const float*)d_in[26];
  const float* rat_bb2 = (const float*)d_in[27];
  const float* gate_W1 = (const float*)d_in[28];
  const float* gate_b1 = (const float*)d_in[29];
  const float* gate_g  = (const float*)d_in[30];
  const float* gate_bb = (const float*)d_in[31];
  const float* gate_W2 = (const float*)d_in[32];
  const float* gate_b2 = (const float*)d_in[33];
  const float* pred_W1 = (const float*)d_in[34];
  const float* pred_b1 = (const float*)d_in[35];
  const float* pred_g  = (const float*)d_in[36];
  const float* pred_bb = (const float*)d_in[37];
  const float* pred_W2 = (const float*)d_in[38];
  const float* pred_b2 = (const float*)d_in[39];
  float* out = (float*)d_out;

  // ---- workspace carve (fp32) -------------------------------------------
  float* p = (float*)d_ws;
  float* h0   = p; p += (size_t)N * D128;   // rat stack runs in place here -> xr
  float* hE   = p; p += (size_t)N * D128;   // enc stack -> h_node
  float* ag   = p; p += (size_t)N * D128;   // aggregator / z / gemm2 out
  float* t1   = p; p += (size_t)N * C256;   // gemm1 out
  float* stats= p; p += 512;
  float* sc   = p; p += 256;
  float* sh   = p; p += 256;
  float* gate = p; p += (size_t)N;
  float* poolA= p; p += (size_t)B * D128;
  float* cnts = p; p += B;
  float* rn   = p; p += B;
  float* en   = p; p += B;
  (void)n_in; (void)ws_size;

  const int totND = N * D128;
  const int gridND = (totND + 255) / 256;
  const int edgeBlocks = (E + 7) / 8;
  const int waveBlocksN = (N + 7) / 8;

  // ---- node encode -------------------------------------------------------
  k_nodeenc<<<gridND, 256, 0, stream>>>(x, node_W, node_b, h0, totND);
  hipMemcpyAsync(hE, h0, (size_t)totND * sizeof(float), hipMemcpyDeviceToDevice, stream);

  // ---- one GIN layer ----------------------------------------------------
  auto gin = [&](float* h, const float* eps, const float* edgeT,
                 const float* W1, const float* b1, const float* g1, const float* bb1,
                 const float* W2, const float* b2, const float* g2, const float* bb2,
                 int doRelu) {
    hipMemsetAsync(ag, 0, (size_t)totND * sizeof(float), stream);
    k_edge<<<edgeBlocks, 256, 0, stream>>>(h, edgeT, src, dst, ea, ag, E);
    k_zbuild<<<gridND, 256, 0, stream>>>(ag, h, eps, totND);
    {
      int tiles = (N / 16) * (C256 / 64);
      k_gemm<0><<<(tiles + 7) / 8, 256, 0, stream>>>(ag, W1, b1, t1, N, D128, C256, nullptr, nullptr);
    }
    hipMemsetAsync(stats, 0, 512 * sizeof(float), stream);
    k_stats<<<256, C256, 0, stream>>>(t1, N, C256, stats);
    k_bnfin<<<1, 256, 0, stream>>>(stats, g1, bb1, N, C256, sc, sh);
    {
      int tiles = (N / 16) * (D128 / 64);
      k_gemm<1><<<(tiles + 7) / 8, 256, 0, stream>>>(t1, W2, b2, ag, N, C256, D128, sc, sh);
    }
    hipMemsetAsync(stats, 0, 512 * sizeof(float), stream);
    k_stats<<<256, D128, 0, stream>>>(ag, N, D128, stats);
    k_bnfin<<<1, 256, 0, stream>>>(stats, g2, bb2, N, D128, sc, sh);
    k_resid<<<gridND, 256, 0, stream>>>(h, ag, sc, sh, doRelu, totND);
  };

  // ---- encoder stack (5 layers) on hE -> h_node -------------------------
  for (int l = 0; l < 5; ++l)
    gin(hE, enc_eps + l, enc_edge + (size_t)l * 4 * D128,
        enc_W1 + (size_t)l * D128 * C256, enc_b1 + (size_t)l * C256,
        enc_g1 + (size_t)l * C256, enc_bb1 + (size_t)l * C256,
        enc_W2 + (size_t)l * C256 * D128, enc_b2 + (size_t)l * D128,
        enc_g2 + (size_t)l * D128, enc_bb2 + (size_t)l * D128, l < 4);

  // ---- rationale stack (2 layers) in place on h0 -> xr ------------------
  for (int l = 0; l < 2; ++l)
    gin(h0, rat_eps + l, rat_edge + (size_t)l * 4 * D128,
        rat_W1 + (size_t)l * D128 * C256, rat_b1 + (size_t)l * C256,
        rat_g1 + (size_t)l * C256, rat_bb1 + (size_t)l * C256,
        rat_W2 + (size_t)l * C256 * D128, rat_b2 + (size_t)l * D128,
        rat_g2 + (size_t)l * D128, rat_bb2 + (size_t)l * D128, l < 1);

  // ---- gate head --------------------------------------------------------
  {
    int tiles = (N / 16) * (C256 / 64);
    k_gemm<0><<<(tiles + 7) / 8, 256, 0, stream>>>(h0, gate_W1, gate_b1, t1, N, D128, C256, nullptr, nullptr);
  }
  hipMemsetAsync(stats, 0, 512 * sizeof(float), stream);
  k_stats<<<256, C256, 0, stream>>>(t1, N, C256, stats);
  k_bnfin<<<1, 256, 0, stream>>>(stats, gate_g, gate_bb, N, C256, sc, sh);
  k_gate<<<waveBlocksN, 256, 0, stream>>>(t1, sc, sh, gate_W2, gate_b2, gumbels, gate, N);

  // ---- pooling ----------------------------------------------------------
  hipMemsetAsync(poolA, 0, ((size_t)B * D128 + 3 * B) * sizeof(float), stream);
  k_pool<<<waveBlocksN, 256, 0, stream>>>(hE, gate, batch, poolA, cnts, rn, en, N);
  k_poolfin<<<(B * D128 + 255) / 256, 256, 0, stream>>>(poolA, cnts, B * D128);

  // ---- prediction head --------------------------------------------------
  {
    int tiles = (B / 16) * (C256 / 64);
    k_gemm<0><<<(tiles + 7) / 8, 256, 0, stream>>>(poolA, pred_W1, pred_b1, t1, B, D128, C256, nullptr, nullptr);
  }
  hipMemsetAsync(stats, 0, 512 * sizeof(float), stream);
  k_stats<<<32, C256, 0, stream>>>(t1, B, C256, stats);
  k_bnfin<<<1, 256, 0, stream>>>(stats, pred_g, pred_bb, B, C256, sc, sh);
  k_head<<<(B * NT + 255) / 256, 256, 0, stream>>>(t1, sc, sh, pred_W2, pred_b2, out, B, C256, NT);

  // ---- reg loss ---------------------------------------------------------
  k_loss<<<1, 256, 0, stream>>>(rn, en, out + (size_t)B * NT, B);
}